// FusedBlock0_68109591380137
// MI455X (gfx1250) — compile-verified
//
#include <hip/hip_runtime.h>
#include <hip/hip_bf16.h>

#define NUM_NODES 10000
#define FEAT_DIM  128
#define NUM_EDGES 640000
#define NUM_CHUNKS (NUM_EDGES / 32)   // 20,000 chunks of 32 contiguous edges

#define BLOCK           256
#define WAVES_PER_BLOCK 8
#define SCATTER_BLOCKS  2048

// ---------------- CDNA5 helpers ----------------

__device__ __forceinline__ void s_wait_asynccnt_0() {
#if __has_builtin(__builtin_amdgcn_s_wait_asynccnt)
  __builtin_amdgcn_s_wait_asynccnt(0);
#else
  asm volatile("s_wait_asynccnt 0x0" ::: "memory");
#endif
}

__device__ __forceinline__ void s_wait_asynccnt_1() {
#if __has_builtin(__builtin_amdgcn_s_wait_asynccnt)
  __builtin_amdgcn_s_wait_asynccnt(1);
#else
  asm volatile("s_wait_asynccnt 0x1" ::: "memory");
#endif
}

// Hardware f32 atomic add to global memory (L2-resident here), relaxed/agent.
__device__ __forceinline__ void atomic_add_f32(float* p, float v) {
  __hip_atomic_fetch_add(p, v, __ATOMIC_RELAXED, __HIP_MEMORY_SCOPE_AGENT);
}

#if __has_builtin(__builtin_amdgcn_global_load_async_to_lds_b128)
#define ASYNC_GATHER 1
typedef int v4i __attribute__((ext_vector_type(4)));
typedef __attribute__((address_space(1))) v4i* g_v4i_ptr;
typedef __attribute__((address_space(3))) v4i* l_v4i_ptr;

// Async-copy one 512B feature row into a per-wave LDS slot:
// lane i moves bytes [16*i, 16*i+16) of the row.
__device__ __forceinline__ void async_row_load(const float* grow, float4* lslot, int lane) {
  float* gp = (float*)grow;  // drop const first, then addrspace+bit cast
  __builtin_amdgcn_global_load_async_to_lds_b128(
      (g_v4i_ptr)(gp + lane * 4),
      (l_v4i_ptr)(lslot + lane),
      /*offset=*/0, /*cpol=*/0);
}
#else
#define ASYNC_GATHER 0
#endif

// ---------------- Kernel: gather + scatter-add, one wave32 per edge ----------------
// Defined FIRST so the disasm snippet shows this (hot) kernel.

__global__ void __launch_bounds__(BLOCK)
scatter_kernel(const float* __restrict__ x, const int* __restrict__ ei,
               float* __restrict__ sum, float* __restrict__ cnt) {
  const int lane        = threadIdx.x & 31;
  const int wave_in_blk = threadIdx.x >> 5;
  const int wave        = blockIdx.x * WAVES_PER_BLOCK + wave_in_blk;
  const int stride      = gridDim.x * WAVES_PER_BLOCK;
  const int* __restrict__ src = ei;
  const int* __restrict__ dst = ei + NUM_EDGES;

#if ASYNC_GATHER
  // Double-buffered per-wave LDS slots: 2 x 32 float4 (512B) per wave = 8KB/block.
  __shared__ float4 tile[WAVES_PER_BLOCK * 2 * 32];
  float4* bufs[2] = { &tile[wave_in_blk * 64], &tile[wave_in_blk * 64 + 32] };

  int c = wave;
  int sv = 0, dv = 0;
  if (c < NUM_CHUNKS) {
    // Coalesced index load: lane i holds edge (c*32 + i)'s src/dst.
    sv = src[c * 32 + lane];
    dv = dst[c * 32 + lane];
    // Prologue: issue async copy of the first edge's row.
    async_row_load(x + (size_t)__shfl(sv, 0) * FEAT_DIM, bufs[0], lane);
  }
  int cur = 0;
  for (; c < NUM_CHUNKS; c += stride) {
    const int cn = c + stride;
    int svn = 0, dvn = 0;
    if (cn < NUM_CHUNKS) {           // preload next chunk's indices (coalesced)
      svn = src[cn * 32 + lane];
      dvn = dst[cn * 32 + lane];
    }
    for (int j = 0; j < 32; ++j) {
      // Issue the next row's async copy into the other buffer, then wait
      // for the current one (loads complete in order -> asynccnt<=1 frees it).
      if (j < 31) {
        async_row_load(x + (size_t)__shfl(sv, j + 1) * FEAT_DIM, bufs[cur ^ 1], lane);
        s_wait_asynccnt_1();
      } else if (cn < NUM_CHUNKS) {
        async_row_load(x + (size_t)__shfl(svn, 0) * FEAT_DIM, bufs[cur ^ 1], lane);
        s_wait_asynccnt_1();
      } else {
        s_wait_asynccnt_0();
      }
      const float4 v = bufs[cur][lane];
      const int d = __shfl(dv, j);
      float* p = sum + (size_t)d * FEAT_DIM + lane * 4;
      atomic_add_f32(p + 0, v.x);
      atomic_add_f32(p + 1, v.y);
      atomic_add_f32(p + 2, v.z);
      atomic_add_f32(p + 3, v.w);
      if (lane == 0) atomic_add_f32(cnt + d, 1.0f);
      cur ^= 1;
    }
    sv = svn;
    dv = dvn;
  }
#else
  for (int c = wave; c < NUM_CHUNKS; c += stride) {
    const int sv = src[c * 32 + lane];
    const int dv = dst[c * 32 + lane];
    for (int j = 0; j < 32; ++j) {
      const int s = __shfl(sv, j);
      const int d = __shfl(dv, j);
      if (j + 1 < 32) {
        __builtin_prefetch(x + (size_t)__shfl(sv, j + 1) * FEAT_DIM + lane * 4, 0, 0);
      }
      const float4* row = (const float4*)(x + (size_t)s * FEAT_DIM);
      const float4 v = row[lane];
      float* p = sum + (size_t)d * FEAT_DIM + lane * 4;
      atomic_add_f32(p + 0, v.x);
      atomic_add_f32(p + 1, v.y);
      atomic_add_f32(p + 2, v.z);
      atomic_add_f32(p + 3, v.w);
      if (lane == 0) atomic_add_f32(cnt + d, 1.0f);
    }
  }
#endif
}

// ---------------- Kernel: zero accumulators + passthrough copy of x ----------------

__global__ void __launch_bounds__(BLOCK)
init_out_kernel(const float* __restrict__ x, float* __restrict__ out) {
  // out layout: [sum_agg 1,280,000][count 10,000][x copy 1,280,000]
  const int ZERO4 = (NUM_NODES * FEAT_DIM + NUM_NODES) / 4; // 322,500 float4
  const int COPY4 = (NUM_NODES * FEAT_DIM) / 4;             // 320,000 float4
  int i = blockIdx.x * blockDim.x + threadIdx.x;
  if (i >= ZERO4 + COPY4) return;
  float4* o4 = (float4*)out;
  if (i < ZERO4) {
    o4[i] = make_float4(0.f, 0.f, 0.f, 0.f);
  } else {
    const float4* x4 = (const float4*)x;
    o4[i] = x4[i - ZERO4];
  }
}

// ---------------- Launch ----------------

extern "C" void kernel_launch(void* const* d_in, const int* in_sizes, int n_in,
                              void* d_out, int out_size, void* d_ws, size_t ws_size,
                              hipStream_t stream) {
  const float* x  = (const float*)d_in[0];
  const int*   ei = (const int*)d_in[1]; // [2, NUM_EDGES]: src row then dst row
  float* out = (float*)d_out;
  float* sum = out;                              // 1,280,000
  float* cnt = out + NUM_NODES * FEAT_DIM;       //    10,000
  (void)in_sizes; (void)n_in; (void)out_size; (void)d_ws; (void)ws_size;

  const int init_total  = (NUM_NODES * FEAT_DIM + NUM_NODES) / 4 + (NUM_NODES * FEAT_DIM) / 4;
  const int init_blocks = (init_total + BLOCK - 1) / BLOCK;
  init_out_kernel<<<init_blocks, BLOCK, 0, stream>>>(x, out);
  scatter_kernel<<<SCATTER_BLOCKS, BLOCK, 0, stream>>>(x, ei, sum, cnt);
}